// MultiHeadAttention_12747462934949
// MI455X (gfx1250) — compile-verified
//
#include <hip/hip_runtime.h>
#include <hip/hip_bf16.h>
#include <math.h>

// Sizes fixed by the reference: B=4, T=1024, C=1024, H=16, HD=64
#define Bb 4
#define Tt 1024
#define Cc 1024
#define Hh 16
#define HDim 64
#define N3 3072
#define LN10000 9.210340371976184f

typedef float v2f __attribute__((ext_vector_type(2)));
typedef float v8f __attribute__((ext_vector_type(8)));
typedef int v4i __attribute__((ext_vector_type(4)));

typedef __attribute__((address_space(1))) void gvoid;
typedef __attribute__((address_space(3))) void lvoid;
typedef __attribute__((address_space(1))) v4i gv4i;
typedef __attribute__((address_space(3))) v4i lv4i;

#if __has_builtin(__builtin_amdgcn_global_load_async_to_lds_b128)
#define HAS_ASYNC_LDS 1
#endif

// 16-byte global -> LDS copy. Async (ASYNCcnt-tracked, no VGPR staging) when
// the gfx1250 builtin is available; plain load/store otherwise.
static __device__ __forceinline__ void async_copy16(const float* g, float* l) {
#ifdef HAS_ASYNC_LDS
  gvoid* gp = (gvoid*)const_cast<float*>(g);  // addrspacecast to global
  lvoid* lp = (lvoid*)l;                      // addrspacecast to LDS
  __builtin_amdgcn_global_load_async_to_lds_b128((gv4i*)gp, (lv4i*)lp, 0, 0);
#else
  *reinterpret_cast<float4*>(l) = *reinterpret_cast<const float4*>(g);
#endif
}

#ifdef HAS_ASYNC_LDS
#if __has_builtin(__builtin_amdgcn_s_wait_asynccnt)
#define WAIT_ASYNC(n) __builtin_amdgcn_s_wait_asynccnt(n)
#else
#define WAIT_ASYNC(n) asm volatile("s_wait_asynccnt %0" ::"n"(n) : "memory")
#endif
#else
#define WAIT_ASYNC(n) ((void)0)
#endif

// D = A(16x4 f32) * B(4x16 f32) + C(16x16 f32) -- full-precision CDNA5 matrix op
static __device__ __forceinline__ v8f wmma4(v2f a, v2f b, v8f c) {
  return __builtin_amdgcn_wmma_f32_16x16x4_f32(false, a, false, b, (short)0, c,
                                               false, false);
}

// ---------------------------------------------------------------------------
// Kernel 1: qkv = x @ W_qkv, fused RoPE on q/k, fold 1/sqrt(HD) into q.
// Tile 128(M) x 128(N), 8 waves; wave = 32x64 (2x4 WMMA accumulators).
// Ping-pong LDS buffers, async global->LDS staging overlapped with WMMA.
// Outputs Q,K,V in [B][H][T][HD] layout.
// ---------------------------------------------------------------------------
__global__ __launch_bounds__(256) void qkv_rope_kernel(
    const float* __restrict__ x, const float* __restrict__ W,
    float* __restrict__ Q, float* __restrict__ K, float* __restrict__ V) {
  __shared__ float As[2][128][20];  // pad 20: bank-clean stride, 16B rows
  __shared__ float Bs[2][16][128];

  const int tid = threadIdx.x;
  const int lane = tid & 31, w = tid >> 5;
  const int half = lane >> 4, l16 = lane & 15;
  const int wm = w >> 1, wn = w & 1;
  const int m0 = blockIdx.y * 128, n0 = blockIdx.x * 128;

  auto issue_tile = [&](int st, int kb) {
#pragma unroll
    for (int i = 0; i < 2; ++i) {  // A tile 128x16
      int f = tid + i * 256;
      int m = f >> 2, kq = (f & 3) << 2;
      async_copy16(&x[(size_t)(m0 + m) * Cc + kb + kq], &As[st][m][kq]);
    }
#pragma unroll
    for (int i = 0; i < 2; ++i) {  // B tile 16x128
      int f = tid + i * 256;
      int k = f >> 5, nq = (f & 31) << 2;
      async_copy16(&W[(size_t)(kb + k) * N3 + n0 + nq], &Bs[st][k][nq]);
    }
  };

  auto load_frags = [&](int st, int ks, v2f a[2], v2f bf[4]) {
    const int k0 = ks * 4 + half * 2;  // A layout: VGPR0=K0/K2, VGPR1=K1/K3
#pragma unroll
    for (int ib = 0; ib < 2; ++ib) {
      const float* p = &As[st][wm * 32 + ib * 16 + l16][k0];
      a[ib].x = p[0];
      a[ib].y = p[1];
    }
#pragma unroll
    for (int jb = 0; jb < 4; ++jb) {
      const int col = wn * 64 + jb * 16 + l16;
      bf[jb].x = Bs[st][k0][col];
      bf[jb].y = Bs[st][k0 + 1][col];
    }
  };

  v8f acc[2][4] = {};

  issue_tile(0, 0);
  for (int it = 0; it < Cc / 16; ++it) {
    if (it + 1 < Cc / 16) {
      issue_tile((it + 1) & 1, (it + 1) * 16);  // overlaps with compute below
      WAIT_ASYNC(4);                            // in-order: tile `it` landed
    } else {
      WAIT_ASYNC(0);
    }
    __syncthreads();
    const int st = it & 1;

    v2f a_c[2], b_c[4];
    load_frags(st, 0, a_c, b_c);
#pragma unroll
    for (int ks = 0; ks < 4; ++ks) {
      v2f a_n[2], b_n[4];
      if (ks < 3) load_frags(st, ks + 1, a_n, b_n);  // hide ds latency
#pragma unroll
      for (int ib = 0; ib < 2; ++ib)
#pragma unroll
        for (int jb = 0; jb < 4; ++jb)
          acc[ib][jb] = wmma4(a_c[ib], b_c[jb], acc[ib][jb]);
      if (ks < 3) {
#pragma unroll
        for (int ib = 0; ib < 2; ++ib) a_c[ib] = a_n[ib];
#pragma unroll
        for (int jb = 0; jb < 4; ++jb) b_c[jb] = b_n[jb];
      }
    }
    __syncthreads();  // buffer st free for reuse at iteration it+2
  }

  // Epilogue: RoPE (interleaved rotate pairs live in adjacent lanes) + scatter.
#pragma unroll
  for (int ib = 0; ib < 2; ++ib) {
#pragma unroll
    for (int jb = 0; jb < 4; ++jb) {
      const int gn = n0 + wn * 64 + jb * 16 + l16;
      const int which = gn >> 10;  // 0=q 1=k 2=v (uniform per wave/jb)
      const int rem = gn & 1023;
      const int hh = rem >> 6;
      const int d = rem & 63;
      float* dst = (which == 0) ? Q : ((which == 1) ? K : V);
#pragma unroll
      for (int r = 0; r < 8; ++r) {
        const int gm = m0 + wm * 32 + ib * 16 + r + half * 8;
        const int bb = gm >> 10, t = gm & 1023;
        float val = acc[ib][jb][r];
        float partner = __shfl_xor(val, 1, 32);  // d ^ 1 neighbor
        float outv;
        if (which < 2) {
          const int j = d & 31;  // emb = concat(freqs,freqs)
          float invf = __expf((float)j * (-2.0f / 64.0f) * LN10000);
          float ang = (float)t * invf;
          float s, c;
          __sincosf(ang, &s, &c);
          float rot = (d & 1) ? partner : -partner;  // rotate_every_two
          outv = val * c + rot * s;
          if (which == 0) outv *= 0.125f;  // fold 1/sqrt(64)
        } else {
          outv = val;
        }
        dst[(((size_t)bb * Hh + hh) * Tt + t) * HDim + d] = outv;
      }
    }
  }
}

// ---------------------------------------------------------------------------
// Kernel 2: flash-style causal attention for one (b,h,64-row q block).
// 4 waves; wave owns 16 q rows. S and O via fp32 WMMA; online softmax with
// half-wave shuffle reductions; P staged through LDS (reusing K tile).
// ---------------------------------------------------------------------------
__global__ __launch_bounds__(128) void attn_kernel(
    const float* __restrict__ Q, const float* __restrict__ K,
    const float* __restrict__ V, const int* __restrict__ amask,
    float* __restrict__ Y) {
  __shared__ float Qs[64][68];   // pad 68: conflict-free strided frag reads
  __shared__ float KPs[64][68];  // K tile, later reused as P staging
  __shared__ float Vs[64 * 64];

  const int tid = threadIdx.x;
  const int lane = tid & 31, w = tid >> 5;
  const int half = lane >> 4, l16 = lane & 15;
  const int qb = blockIdx.x;
  const int bh = blockIdx.y;
  const int b = bh >> 4, h = bh & 15;
  const size_t headoff = ((size_t)b * Hh + h) * Tt * HDim;

#pragma unroll
  for (int i = 0; i < 8; ++i) {  // Q tile 64x64 (async)
    int f = tid + i * 128;
    int row = f >> 4, dq = (f & 15) << 2;
    async_copy16(&Q[headoff + (size_t)(qb * 64 + row) * HDim + dq],
                 &Qs[row][dq]);
  }

  v8f o[4] = {};
  float mrow[8], lrow[8];
#pragma unroll
  for (int r = 0; r < 8; ++r) {
    mrow[r] = -3.0e38f;
    lrow[r] = 0.0f;
  }

  for (int kb = 0; kb <= qb; ++kb) {
    __syncthreads();  // previous P/V reads complete before overwrite
#pragma unroll
    for (int i = 0; i < 8; ++i) {  // K,V tiles 64x64 each (async)
      int f = tid + i * 128;
      int row = f >> 4, dq = (f & 15) << 2;
      async_copy16(&K[headoff + (size_t)(kb * 64 + row) * HDim + dq],
                   &KPs[row][dq]);
      async_copy16(&V[headoff + (size_t)(kb * 64 + row) * HDim + dq],
                   &Vs[row * 64 + dq]);
    }
    WAIT_ASYNC(0);  // all async tile data landed in LDS
    __syncthreads();

    // S(16x64) = Q_rows(16x64) @ K^T(64x64): B[kdim][n] = KPs[n][kdim]
    v8f s[4] = {};
#pragma unroll
    for (int ks = 0; ks < 16; ++ks) {
      const int k0 = ks * 4 + half * 2;
      v2f a;
      {
        const float* p = &Qs[w * 16 + l16][k0];
        a.x = p[0];
        a.y = p[1];
      }
#pragma unroll
      for (int jb = 0; jb < 4; ++jb) {
        v2f bf;
        const int col = jb * 16 + l16;
        bf.x = KPs[col][k0];
        bf.y = KPs[col][k0 + 1];
        s[jb] = wmma4(a, bf, s[jb]);
      }
    }

    // causal mask (diagonal block only) + pad mask
    const bool diag = (kb == qb);
    int pm[4];
#pragma unroll
    for (int jb = 0; jb < 4; ++jb)
      pm[jb] = amask[b * Tt + kb * 64 + jb * 16 + l16];
#pragma unroll
    for (int jb = 0; jb < 4; ++jb) {
      const int kg = kb * 64 + jb * 16 + l16;
#pragma unroll
      for (int r = 0; r < 8; ++r) {
        const int qg = qb * 64 + w * 16 + r + half * 8;
        if ((diag && kg > qg) || pm[jb] == 0) s[jb][r] = -3.0e38f;
      }
    }

    // online softmax: row lives across 16 lanes of one wave half
#pragma unroll
    for (int r = 0; r < 8; ++r) {
      float mx = fmaxf(fmaxf(s[0][r], s[1][r]), fmaxf(s[2][r], s[3][r]));
      mx = fmaxf(mx, __shfl_xor(mx, 1, 16));
      mx = fmaxf(mx, __shfl_xor(mx, 2, 16));
      mx = fmaxf(mx, __shfl_xor(mx, 4, 16));
      mx = fmaxf(mx, __shfl_xor(mx, 8, 16));
      const float mnew = fmaxf(mrow[r], mx);
      const float sf = __expf(mrow[r] - mnew);
      mrow[r] = mnew;
      float rs = 0.0f;
#pragma unroll
      for (int jb = 0; jb < 4; ++jb) {
        const float p = __expf(s[jb][r] - mnew);
        s[jb][r] = p;
        rs += p;
      }
      rs += __shfl_xor(rs, 1, 16);
      rs += __shfl_xor(rs, 2, 16);
      rs += __shfl_xor(rs, 4, 16);
      rs += __shfl_xor(rs, 8, 16);
      lrow[r] = lrow[r] * sf + rs;
#pragma unroll
      for (int jb = 0; jb < 4; ++jb) o[jb][r] *= sf;
    }

    __syncthreads();  // all waves done reading K before P overwrites KPs

    // stage P (C-layout -> A-layout) into own 16 rows of KPs
#pragma unroll
    for (int jb = 0; jb < 4; ++jb)
#pragma unroll
      for (int r = 0; r < 8; ++r)
        KPs[w * 16 + r + half * 8][jb * 16 + l16] = s[jb][r];

    // O(16x64) += P(16x64) @ V(64x64)
#pragma unroll
    for (int ks = 0; ks < 16; ++ks) {
      const int k0 = ks * 4 + half * 2;
      v2f a;
      {
        const float* p = &KPs[w * 16 + l16][k0];
        a.x = p[0];
        a.y = p[1];
      }
#pragma unroll
      for (int jb = 0; jb < 4; ++jb) {
        v2f bf;
        bf.x = Vs[k0 * 64 + jb * 16 + l16];
        bf.y = Vs[(k0 + 1) * 64 + jb * 16 + l16];
        o[jb] = wmma4(a, bf, o[jb]);
      }
    }
  }

  // normalize, store into (B,T,C) layout for the projection GEMM
#pragma unroll
  for (int r = 0; r < 8; ++r) {
    const float inv = 1.0f / lrow[r];
    const int t = qb * 64 + w * 16 + r + half * 8;
#pragma unroll
    for (int jb = 0; jb < 4; ++jb)
      Y[((size_t)b * Tt + t) * Cc + h * HDim + jb * 16 + l16] = o[jb][r] * inv;
  }
}

// ---------------------------------------------------------------------------
// Kernel 3: out = Y @ W_proj + b_proj. Same pipelined tiling as kernel 1.
// ---------------------------------------------------------------------------
__global__ __launch_bounds__(256) void proj_kernel(
    const float* __restrict__ Yt, const float* __restrict__ W,
    const float* __restrict__ bias, float* __restrict__ Out) {
  __shared__ float As[2][128][20];
  __shared__ float Bs[2][16][128];

  const int tid = threadIdx.x;
  const int lane = tid & 31, w = tid >> 5;
  const int half = lane >> 4, l16 = lane & 15;
  const int wm = w >> 1, wn = w & 1;
  const int m0 = blockIdx.y * 128, n0 = blockIdx.x * 128;

  auto issue_tile = [&](int st, int kb) {
#pragma unroll
    for (int i = 0; i < 2; ++i) {
      int f = tid + i * 256;
      int m = f >> 2, kq = (f & 3) << 2;
      async_copy16(&Yt[(size_t)(m0 + m) * Cc + kb + kq], &As[st][m][kq]);
    }
#pragma unroll
    for (int i = 0; i < 2; ++i) {
      int f = tid + i * 256;
      int k = f >> 5, nq = (f & 31) << 2;
      async_copy16(&W[(size_t)(kb + k) * Cc + n0 + nq], &Bs[st][k][nq]);
    }
  };

  auto load_frags = [&](int st, int ks, v2f a[2], v2f bf[4]) {
    const int k0 = ks * 4 + half * 2;
#pragma unroll
    for (int ib = 0; ib < 2; ++ib) {
      const float* p = &As[st][wm * 32 + ib * 16 + l16][k0];
      a[ib].x = p[0];
      a[ib].y = p[1];
    }
#pragma unroll
    for (int jb = 0; jb < 4; ++jb) {
      const int col = wn * 64 + jb * 16 + l16;
      bf[jb].x = Bs[st][k0][col];
      bf[jb].y = Bs[st][k0 + 1][col];
    }
  };

  v8f acc[2][4] = {};

  issue_tile(0, 0);
  for (int it = 0; it < Cc / 16; ++it) {
    if (it + 1 < Cc / 16) {
      issue_tile((it + 1) & 1, (it + 1) * 16);
      WAIT_ASYNC(4);
    } else {
      WAIT_ASYNC(0);
    }
    __syncthreads();
    const int st = it & 1;

    v2f a_c[2], b_c[4];
    load_frags(st, 0, a_c, b_c);
#pragma unroll
    for (int ks = 0; ks < 4; ++ks) {
      v2f a_n[2], b_n[4];
      if (ks < 3) load_frags(st, ks + 1, a_n, b_n);
#pragma unroll
      for (int ib = 0; ib < 2; ++ib)
#pragma unroll
        for (int jb = 0; jb < 4; ++jb)
          acc[ib][jb] = wmma4(a_c[ib], b_c[jb], acc[ib][jb]);
      if (ks < 3) {
#pragma unroll
        for (int ib = 0; ib < 2; ++ib) a_c[ib] = a_n[ib];
#pragma unroll
        for (int jb = 0; jb < 4; ++jb) b_c[jb] = b_n[jb];
      }
    }
    __syncthreads();
  }

#pragma unroll
  for (int ib = 0; ib < 2; ++ib)
#pragma unroll
    for (int jb = 0; jb < 4; ++jb) {
      const int gn = n0 + wn * 64 + jb * 16 + l16;
      const float bv = bias[gn];
#pragma unroll
      for (int r = 0; r < 8; ++r) {
        const int gm = m0 + wm * 32 + ib * 16 + r + half * 8;
        Out[(size_t)gm * Cc + gn] = acc[ib][jb][r] + bv;
      }
    }
}

// ---------------------------------------------------------------------------
extern "C" void kernel_launch(void* const* d_in, const int* in_sizes, int n_in,
                              void* d_out, int out_size, void* d_ws,
                              size_t ws_size, hipStream_t stream) {
  (void)in_sizes;
  (void)n_in;
  (void)out_size;
  (void)ws_size;
  const float* x = (const float*)d_in[0];
  const int* amask = (const int*)d_in[1];
  const float* Wqkv = (const float*)d_in[2];
  const float* Wproj = (const float*)d_in[3];
  const float* bproj = (const float*)d_in[4];
  float* out = (float*)d_out;

  const size_t SZ = (size_t)Bb * Hh * Tt * HDim;  // 4,194,304 floats
  float* Q = (float*)d_ws;
  float* K = Q + SZ;
  float* V = K + SZ;
  float* Yt = V + SZ;

  qkv_rope_kernel<<<dim3(N3 / 128, (Bb * Tt) / 128), 256, 0, stream>>>(
      x, Wqkv, Q, K, V);
  attn_kernel<<<dim3(Tt / 64, Bb * Hh), 128, 0, stream>>>(Q, K, V, amask, Yt);
  proj_kernel<<<dim3(Cc / 128, (Bb * Tt) / 128), 256, 0, stream>>>(
      Yt, Wproj, bproj, out);
}